// TopologyLayer_72945724555270
// MI455X (gfx1250) — compile-verified
//
#include <hip/hip_runtime.h>
#include <math.h>

typedef float v2f __attribute__((ext_vector_type(2)));
typedef float v8f __attribute__((ext_vector_type(8)));

#define F_IN 128
#define HID  64
#define NF   8
#define NCF  9          // 3 tri + 3 gauss + 3 line
#define KACT (NF*NCF)   // 72
#define KTOT (F_IN+KACT)// 200
#define MT   128        // rows per workgroup tile
#define INV2SIG2 50.0f  // 1/(2*0.1^2)

static __device__ __forceinline__ v8f wmma4(v2f a, v2f b, v8f c) {
    // D = A(16x4,f32) * B(4x16,f32) + C(16x16,f32)
    return __builtin_amdgcn_wmma_f32_16x16x4_f32(false, a, false, b, (short)0, c, false, false);
}

static __device__ __forceinline__ float coordfun(float p0, float p1, int c,
                                                 const float* tt, const float* gt,
                                                 const float* lw, const float* lb) {
    if (c < 3) {                       // Triangle_transform
        return fmaxf(p1 - fabsf(tt[c] - p0), 0.0f);
    } else if (c < 6) {                // Gaussian_transform
        int i = c - 3;
        float dx = p0 - gt[2*i], dy = p1 - gt[2*i+1];
        return expf(-(dx*dx + dy*dy) * INV2SIG2);
    } else {                           // Line_transform
        int i = c - 6;
        return p0 * lw[i] + p1 * lw[3+i] + lb[i];
    }
}

// ---------------------------------------------------------------------------
// Kernel 1: fused  H=relu(x@W1+b1)  ->  fv=H@W2+b2  ->  act0  ->
//           out=relu([x|act0]@out_W+out_b).  128 rows / block, 8 waves.
// ---------------------------------------------------------------------------
__global__ __launch_bounds__(256, 1)
void fused_node_kernel(const float* __restrict__ x,
                       const float* __restrict__ W1, const float* __restrict__ b1,
                       const float* __restrict__ W2, const float* __restrict__ b2,
                       const float* __restrict__ tri_t, const float* __restrict__ gauss_t,
                       const float* __restrict__ line_W, const float* __restrict__ line_b,
                       const float* __restrict__ outW, const float* __restrict__ outb,
                       float* __restrict__ fvg, float* __restrict__ outp) {
    __shared__ float sW1[F_IN][HID + 1];                  //  8320 f
    __shared__ float sW2[HID][17];                        //  1088 f (cols 8..15 zeroed)
    __shared__ float sOW[KTOT][F_IN + 1];                 // 25800 f
    __shared__ __align__(16) float sX[MT][F_IN + 1];      // 16512 f (reused as out stage)
    __shared__ float sScr[MT * (KACT + 1)];               //  9344 f: H[..][65] / Act[..][73]
    __shared__ float sFv[MT][NF + 1];                     //  1152 f

    const int tid  = threadIdx.x;
    const int lane = tid & 31;
    const int wave = tid >> 5;
    const int rl   = lane & 15;      // row within 16-row fragment
    const int hi   = lane >> 4;      // half-wave: K offset +2 per ISA layout
    const int rbase = wave * 16;     // this wave's rows within tile
    const int row0  = blockIdx.x * MT;

    // ---- cooperative loads into LDS (b128 on the global side) ----
    {
        const float4* W1v = (const float4*)W1;
        for (int i = tid; i < (F_IN * HID) / 4; i += 256) {
            float4 v = W1v[i];
            int base = i * 4, r = base / HID, c = base % HID;
            sW1[r][c] = v.x; sW1[r][c+1] = v.y; sW1[r][c+2] = v.z; sW1[r][c+3] = v.w;
        }
        const float4* OWv = (const float4*)outW;
        for (int i = tid; i < (KTOT * F_IN) / 4; i += 256) {
            float4 v = OWv[i];
            int base = i * 4, r = base / F_IN, c = base % F_IN;
            sOW[r][c] = v.x; sOW[r][c+1] = v.y; sOW[r][c+2] = v.z; sOW[r][c+3] = v.w;
        }
        const float4* Xv = (const float4*)(x + (size_t)row0 * F_IN);
        for (int i = tid; i < (MT * F_IN) / 4; i += 256) {
            float4 v = Xv[i];
            int base = i * 4, r = base / F_IN, c = base % F_IN;
            sX[r][c] = v.x; sX[r][c+1] = v.y; sX[r][c+2] = v.z; sX[r][c+3] = v.w;
        }
        for (int i = tid; i < HID * 16; i += 256) {
            int k = i >> 4, n = i & 15;
            sW2[k][n] = (n < NF) ? W2[k * NF + n] : 0.0f;
        }
    }
    __syncthreads();

    // ---- GEMM1: H = relu(x @ W1 + b1), M=16/wave, N=64, K=128 ----
    v8f c1[4];
    #pragma unroll
    for (int t = 0; t < 4; ++t) c1[t] = (v8f){0,0,0,0,0,0,0,0};
    for (int kk = 0; kk < F_IN / 4; ++kk) {
        int k = kk * 4 + hi * 2;
        v2f a; a.x = sX[rbase + rl][k]; a.y = sX[rbase + rl][k + 1];
        #pragma unroll
        for (int t = 0; t < 4; ++t) {
            int n = t * 16 + rl;
            v2f b; b.x = sW1[k][n]; b.y = sW1[k + 1][n];
            c1[t] = wmma4(a, b, c1[t]);
        }
    }
    float* sH = sScr;   // H[MT][HID+1]
    #pragma unroll
    for (int t = 0; t < 4; ++t) {
        int n = t * 16 + rl;
        float bias = b1[n];
        #pragma unroll
        for (int v = 0; v < 8; ++v) {
            int r = rbase + v + 8 * hi;
            sH[r * (HID + 1) + n] = fmaxf(c1[t][v] + bias, 0.0f);
        }
    }
    // same-wave LDS ops are in-order; GEMM2 reads only this wave's rows

    // ---- GEMM2: fv = H @ W2 + b2, N=16 (8 valid), K=64 ----
    v8f c2 = (v8f){0,0,0,0,0,0,0,0};
    for (int kk = 0; kk < HID / 4; ++kk) {
        int k = kk * 4 + hi * 2;
        v2f a; a.x = sH[(rbase + rl) * (HID + 1) + k];
               a.y = sH[(rbase + rl) * (HID + 1) + k + 1];
        v2f b; b.x = sW2[k][rl]; b.y = sW2[k + 1][rl];
        c2 = wmma4(a, b, c2);
    }
    __syncthreads();    // everyone done reading H before Act overwrites sScr

    if (rl < NF) {
        float bias = b2[rl];
        #pragma unroll
        for (int v = 0; v < 8; ++v) {
            int r = rbase + v + 8 * hi;
            float f = c2[v] + bias;
            sFv[r][rl] = f;
            fvg[(size_t)(row0 + r) * NF + rl] = f;
        }
    }

    // ---- act0 for this wave's 16 rows (p0=p1=fv) ----
    float* sAct = sScr; // Act[MT][KACT+1]
    for (int e = lane; e < 16 * KACT; e += 32) {
        int r = rbase + e / KACT;
        int j = e % KACT;
        int f = j / NCF, c = j % NCF;
        float v = sFv[r][f];
        sAct[r * (KACT + 1) + j] = coordfun(v, v, c, tri_t, gauss_t, line_W, line_b);
    }

    // ---- GEMM3: out = relu([x | act0] @ out_W + out_b), N=128, K=200 ----
    v8f c3[8];
    #pragma unroll
    for (int t = 0; t < 8; ++t) c3[t] = (v8f){0,0,0,0,0,0,0,0};
    for (int kk = 0; kk < F_IN / 4; ++kk) {
        int k = kk * 4 + hi * 2;
        v2f a; a.x = sX[rbase + rl][k]; a.y = sX[rbase + rl][k + 1];
        #pragma unroll
        for (int t = 0; t < 8; ++t) {
            int n = t * 16 + rl;
            v2f b; b.x = sOW[k][n]; b.y = sOW[k + 1][n];
            c3[t] = wmma4(a, b, c3[t]);
        }
    }
    for (int kk = 0; kk < KACT / 4; ++kk) {
        int k = kk * 4 + hi * 2;
        v2f a; a.x = sAct[(rbase + rl) * (KACT + 1) + k];
               a.y = sAct[(rbase + rl) * (KACT + 1) + k + 1];
        int kg = F_IN + k;
        #pragma unroll
        for (int t = 0; t < 8; ++t) {
            int n = t * 16 + rl;
            v2f b; b.x = sOW[kg][n]; b.y = sOW[kg + 1][n];
            c3[t] = wmma4(a, b, c3[t]);
        }
    }

    // ---- epilogue: bias+relu -> stage in LDS (reuse sX, stride F_IN) ->
    //      cooperative contiguous b128 global stores ----
    __syncthreads();                    // all waves done reading sX
    float* sOut = &sX[0][0];            // [MT*F_IN], 16B-aligned
    #pragma unroll
    for (int t = 0; t < 8; ++t) {
        int n = t * 16 + rl;
        float bias = outb[n];
        #pragma unroll
        for (int v = 0; v < 8; ++v) {
            int r = rbase + v + 8 * hi;
            sOut[r * F_IN + n] = fmaxf(c3[t][v] + bias, 0.0f);
        }
    }
    __syncthreads();
    {
        float4* outv = (float4*)(outp + (size_t)row0 * F_IN);
        const float4* sOutv = (const float4*)sOut;
        for (int i = tid; i < (MT * F_IN) / 4; i += 256)
            outv[i] = sOutv[i];
    }
}

// ---------------------------------------------------------------------------
// Kernel 2: unpaired[g,f] = max over graph g's contiguous nodes of fv[:,f]
// ---------------------------------------------------------------------------
__global__ void seg_max_kernel(const float* __restrict__ fvg,
                               float* __restrict__ unpaired, int nodes_per_g) {
    int g = blockIdx.x;
    int f = threadIdx.x >> 5;
    int lane = threadIdx.x & 31;
    float m = -INFINITY;
    for (int i = lane; i < nodes_per_g; i += 32)
        m = fmaxf(m, fvg[(size_t)(g * nodes_per_g + i) * NF + f]);
    #pragma unroll
    for (int off = 16; off > 0; off >>= 1)
        m = fmaxf(m, __shfl_xor(m, off, 32));
    if (lane == 0) unpaired[g * NF + f] = m;
}

// ---------------------------------------------------------------------------
// Kernel 3: analytic dim-1 branch.  Only the random edges are masked, so
// g1[g,f,c] = (mask_f ? act(vals[g,f]) : 0) + (Mg - mask_f) * act((0,0)).
// ---------------------------------------------------------------------------
__global__ void edge_g1_kernel(const float* __restrict__ fvg,
                               const float* __restrict__ unpaired,
                               const int* __restrict__ edge_index,
                               const int* __restrict__ random_edges,
                               const float* __restrict__ tri_t1,
                               const float* __restrict__ gauss_t1,
                               const float* __restrict__ line_W1,
                               const float* __restrict__ line_b1,
                               float* __restrict__ g1, int E) {
    __shared__ float2 sVals[NF];
    __shared__ int sEdge[NF], sNz[NF], sMask[NF], sMg;
    int g = blockIdx.x, t = threadIdx.x;
    if (t < NF) {
        int e = random_edges[g * NF + t];
        int u = edge_index[e], v = edge_index[(size_t)E + e];
        float fe = fmaxf(fvg[(size_t)u * NF + t], fvg[(size_t)v * NF + t]);
        float up = unpaired[g * NF + t];
        sVals[t] = make_float2(up, fe);
        sEdge[t] = e;
        sNz[t] = (up != 0.0f) || (fe != 0.0f);
    }
    __syncthreads();
    if (t == 0) {
        int mg = 0;
        for (int f = 0; f < NF; ++f) {
            int msk = 0, first = 1;
            for (int f2 = 0; f2 < NF; ++f2)
                if (sEdge[f2] == sEdge[f] && sNz[f2]) msk = 1;
            for (int f2 = 0; f2 < f; ++f2)
                if (sEdge[f2] == sEdge[f]) first = 0;
            sMask[f] = msk;
            if (msk && first) ++mg;
        }
        sMg = mg;
    }
    __syncthreads();
    if (t < KACT) {
        int f = t / NCF, c = t % NCF;
        float a    = coordfun(sVals[f].x, sVals[f].y, c, tri_t1, gauss_t1, line_W1, line_b1);
        float base = coordfun(0.0f, 0.0f, c, tri_t1, gauss_t1, line_W1, line_b1);
        int m = sMask[f];
        g1[(size_t)g * KACT + t] = (m ? a : 0.0f) + (float)(sMg - m) * base;
    }
}

// ---------------------------------------------------------------------------
extern "C" void kernel_launch(void* const* d_in, const int* in_sizes, int n_in,
                              void* d_out, int out_size, void* d_ws, size_t ws_size,
                              hipStream_t stream) {
    const float* x        = (const float*)d_in[0];
    const int*   edge_idx = (const int*)  d_in[1];
    const int*   rnd_e    = (const int*)  d_in[4];
    const float* W1       = (const float*)d_in[5];
    const float* b1       = (const float*)d_in[6];
    const float* W2       = (const float*)d_in[7];
    const float* b2       = (const float*)d_in[8];
    const float* tri_t    = (const float*)d_in[9];
    const float* gauss_t  = (const float*)d_in[10];
    const float* line_W   = (const float*)d_in[11];
    const float* line_b   = (const float*)d_in[12];
    const float* tri_t1   = (const float*)d_in[13];
    const float* gauss_t1 = (const float*)d_in[14];
    const float* line_W1  = (const float*)d_in[15];
    const float* line_b1  = (const float*)d_in[16];
    const float* out_W    = (const float*)d_in[17];
    const float* out_b    = (const float*)d_in[18];
    float* outp = (float*)d_out;

    const int N  = in_sizes[0] / F_IN;      // 204800
    const int E  = in_sizes[1] / 2;         // 1638400
    const int BS = in_sizes[4] / NF;        // 256
    const int nodes_per_g = N / BS;         // 800

    float* fvg      = (float*)d_ws;                 // [N, 8]
    float* unpaired = fvg + (size_t)N * NF;         // [BS, 8]

    fused_node_kernel<<<N / MT, 256, 0, stream>>>(
        x, W1, b1, W2, b2, tri_t, gauss_t, line_W, line_b, out_W, out_b, fvg, outp);
    seg_max_kernel<<<BS, 256, 0, stream>>>(fvg, unpaired, nodes_per_g);
    edge_g1_kernel<<<BS, 96, 0, stream>>>(
        fvg, unpaired, edge_idx, rnd_e, tri_t1, gauss_t1, line_W1, line_b1,
        outp + (size_t)N * F_IN, E);
}